// AGNN_27908697489542
// MI455X (gfx1250) — compile-verified
//
#include <hip/hip_runtime.h>
#include <math.h>

// AGNN propagation for MI455X (gfx1250, wave32).
// Phases: 1) per-node inv-norm + init, 2) per-edge cosine logits via
// V_WMMA_F32_16X16X4_F32 on RAW feature rows (norm scaling folded into the
// diagonal extraction: D[m][m] = dn_m*sn_m*dot_raw), 3) exp + segment-sum,
// 4) alpha-weighted scatter-add of raw source features.

typedef __attribute__((ext_vector_type(2))) float v2f;
typedef __attribute__((ext_vector_type(8))) float v8f;

#define DF 32
#define AGNN_EPS 1e-8f

// Order-preserving float <-> uint encoding so atomicMax(u32) == float max.
__device__ __forceinline__ unsigned f2ord(float f) {
  unsigned u = __float_as_uint(f);
  return (u & 0x80000000u) ? ~u : (u | 0x80000000u);
}
__device__ __forceinline__ float ord2f(unsigned u) {
  return (u & 0x80000000u) ? __uint_as_float(u & 0x7fffffffu)
                           : __uint_as_float(~u);
}

// ---- Phase 1: inv_norm per node, zero segmax/segsum/out -------------------
__global__ void agnn_prep(const float* __restrict__ x,
                          float* __restrict__ inv_norm,
                          unsigned* __restrict__ segmax,
                          float* __restrict__ segsum,
                          float* __restrict__ out, int n_nodes) {
  int i = blockIdx.x * blockDim.x + threadIdx.x;
  if (i >= n_nodes) return;
  const float4* row = (const float4*)(x + (size_t)i * DF);
  float s = 0.f;
#pragma unroll
  for (int k = 0; k < DF / 4; ++k) {
    float4 v = row[k];
    s += v.x * v.x + v.y * v.y + v.z * v.z + v.w * v.w;
  }
  inv_norm[i] = rsqrtf(s + AGNN_EPS);
  segmax[i] = 0u;   // minimum of the ordered-uint encoding
  segsum[i] = 0.f;
  float4* orow = (float4*)(out + (size_t)i * DF);
  float4 z = make_float4(0.f, 0.f, 0.f, 0.f);
#pragma unroll
  for (int k = 0; k < DF / 4; ++k) orow[k] = z;
}

// ---- Phase 2: 16 edge dots per wave via WMMA f32 16x16x4 ------------------
// A (16x32) = raw dst rows, B (32x16) = raw src rows as columns; accumulate
// 8 K-chunks of 4. Diagonal of the 16x16 result = raw dot(dst_e, src_e);
// cosine = dn*sn*raw_dot, folded into one scalar multiply per edge since the
// extracting lane (0-7 / 24-31) owns exactly that edge's dn,sn.
__global__ void agnn_edge_logits(const float* __restrict__ x,
                                 const float* __restrict__ beta,
                                 const int* __restrict__ srcs,
                                 const int* __restrict__ dsts,
                                 const float* __restrict__ inv_norm,
                                 float* __restrict__ logits,
                                 unsigned* __restrict__ segmax,
                                 int n_edges) {
  const int lane = (int)(threadIdx.x & 31u);
  const unsigned gtid = blockIdx.x * blockDim.x + threadIdx.x;
  const int wave = (int)(gtid >> 5);
  const int e0 = wave * 16;
  if (e0 >= n_edges) return;  // wave-uniform: EXEC stays all-ones for WMMA

  const int half = lane >> 4;   // 0: K pair {k0,k0+1}, 1: K pair {k0+2,k0+3}
  const int m = lane & 15;      // row (A) / column (B) index == edge slot
  int e = e0 + m;
  if (e >= n_edges) e = n_edges - 1;  // clamp loads for a partial tail wave
  const int di = dsts[e];
  const int si = srcs[e];
  const float dn = inv_norm[di];
  const float sn = inv_norm[si];
  const float* __restrict__ drow = x + (size_t)di * DF;
  const float* __restrict__ srow = x + (size_t)si * DF;

  v8f c = {};
#pragma unroll
  for (int kc = 0; kc < 8; ++kc) {
    const int k = kc * 4 + half * 2;
    const v2f a = *(const v2f*)(drow + k);  // 8B-aligned global_load_b64
    const v2f b = *(const v2f*)(srow + k);
    // D = A(16x4) * B(4x16) + C
    c = __builtin_amdgcn_wmma_f32_16x16x4_f32(
        /*neg_a=*/false, a, /*neg_b=*/false, b,
        /*c_mod=*/(short)0, c, /*reuse_a=*/false, /*reuse_b=*/false);
  }

  // Diagonal extraction from the 16x16 f32 C/D layout:
  //   edge d<8  -> VGPR d,   lane d
  //   edge d>=8 -> VGPR d-8, lane d+16
  const bool active = (lane < 8) || (lane >= 24);
  const int vidx = lane & 7;
  float dot = c[0];
#pragma unroll
  for (int i = 1; i < 8; ++i) dot = (vidx == i) ? c[i] : dot;

  const int eout = e0 + ((lane < 8) ? lane : (lane - 16));
  if (active && eout < n_edges) {
    const float logit = beta[0] * (dot * dn * sn);  // cosine * beta
    logits[eout] = logit;
    atomicMax(&segmax[dsts[eout]], f2ord(logit));
  }
}

// ---- Phase 3: e = exp(logit - segmax[dst]); segsum[dst] += e --------------
__global__ void agnn_exp_sum(const int* __restrict__ dsts,
                             const unsigned* __restrict__ segmax,
                             float* __restrict__ logits,
                             float* __restrict__ segsum, int n_edges) {
  int e = blockIdx.x * blockDim.x + threadIdx.x;
  if (e >= n_edges) return;
  const int d = dsts[e];
  const float mx = ord2f(segmax[d]);
  const float ex = expf(logits[e] - mx);
  logits[e] = ex;  // reuse buffer for exp values
  atomicAdd(&segsum[d], ex);
}

// ---- Phase 4: out[dst] += alpha * x[src] ----------------------------------
__global__ void agnn_aggregate(const float* __restrict__ x,
                               const int* __restrict__ srcs,
                               const int* __restrict__ dsts,
                               const float* __restrict__ evals,
                               const float* __restrict__ segsum,
                               float* __restrict__ out, int n_edges) {
  int e = blockIdx.x * blockDim.x + threadIdx.x;
  if (e >= n_edges) return;
  const int d = dsts[e];
  const int s = srcs[e];
  const float alpha = evals[e] / (segsum[d] + AGNN_EPS);
  const float* __restrict__ srow = x + (size_t)s * DF;
  float* __restrict__ orow = out + (size_t)d * DF;
#pragma unroll
  for (int k = 0; k < DF; ++k) atomicAdd(&orow[k], alpha * srow[k]);
}

extern "C" void kernel_launch(void* const* d_in, const int* in_sizes, int n_in,
                              void* d_out, int out_size, void* d_ws,
                              size_t ws_size, hipStream_t stream) {
  const float* x = (const float*)d_in[0];     // [N, 32] f32
  const float* beta = (const float*)d_in[1];  // [1] f32
  const int* ei = (const int*)d_in[2];        // [2, E] int
  const int n_nodes = in_sizes[0] / DF;
  const int n_edges = in_sizes[2] / 2;
  const int* srcs = ei;             // edge_index[0]
  const int* dsts = ei + n_edges;   // edge_index[1]
  float* out = (float*)d_out;

  // workspace: inv_norm[N] | segmax[N] (u32) | segsum[N] | logits/e[E]
  float* inv_norm = (float*)d_ws;
  unsigned* segmax = (unsigned*)(inv_norm + n_nodes);
  float* segsum = (float*)(segmax + n_nodes);
  float* logits = segsum + n_nodes;

  dim3 blk(256);
  agnn_prep<<<(n_nodes + 255) / 256, blk, 0, stream>>>(x, inv_norm, segmax,
                                                       segsum, out, n_nodes);
  const int nwaves = (n_edges + 15) / 16;          // 16 edges per wave
  const int nblk = (nwaves * 32 + 255) / 256;      // 8 waves per block
  agnn_edge_logits<<<nblk, blk, 0, stream>>>(x, beta, srcs, dsts, inv_norm,
                                             logits, segmax, n_edges);
  agnn_exp_sum<<<(n_edges + 255) / 256, blk, 0, stream>>>(dsts, segmax, logits,
                                                          segsum, n_edges);
  agnn_aggregate<<<(n_edges + 255) / 256, blk, 0, stream>>>(
      x, srcs, dsts, logits, segsum, out, n_edges);
}